// Mamba_48661979464259
// MI455X (gfx1250) — compile-verified
//
#include <hip/hip_runtime.h>
#include <math.h>

typedef __attribute__((ext_vector_type(2))) float v2f;
typedef __attribute__((ext_vector_type(8))) float v8f;
typedef __attribute__((ext_vector_type(4))) unsigned int v4u;
typedef __attribute__((ext_vector_type(8))) unsigned int v8u;

#define BM_SEQ  112      // B*M
#define NPAT    64       // NPATCH
#define DMODEL  256
#define DINNER  512
#define DSTATE  16
#define DTRANK  16
#define DCONV   4
#define PATCHL  16
#define STRIDEL 8
#define SEQLEN  512
#define NROWS   (BM_SEQ*NPAT)   // 7168
#define PREDL   96
#define EPSV    1e-5f

__device__ __forceinline__ float silu_f(float x) { return x / (1.f + expf(-x)); }

__device__ __forceinline__ v8f wmma4(v2f a, v2f b, v8f c) {
  // V_WMMA_F32_16X16X4_F32: (neg_a, A, neg_b, B, c_mod, C, reuse_a, reuse_b)
  return __builtin_amdgcn_wmma_f32_16x16x4_f32(false, a, false, b, (short)0, c, false, false);
}

// ---------------- patch embedding: patches @ pos_W + pos_b + pos_emb ----------------
__global__ void patch_embed(const float* __restrict__ x,
                            const float* __restrict__ posW,
                            const float* __restrict__ posb,
                            const float* __restrict__ pose,
                            float* __restrict__ h)
{
  const int seq = blockIdx.x;   // 0..111 (b*7+m)
  const int t   = blockIdx.y;   // 0..63
  const int d   = threadIdx.x;  // 0..255
  __shared__ float patch[PATCHL];
  if (threadIdx.x < PATCHL) {
    int pos = t * STRIDEL + (int)threadIdx.x;
    if (pos > SEQLEN - 1) pos = SEQLEN - 1;   // padded region repeats last sample
    patch[threadIdx.x] = x[(size_t)seq * SEQLEN + pos];
  }
  __syncthreads();
  float s = posb[d] + pose[(size_t)t * DMODEL + d];
#pragma unroll
  for (int p = 0; p < PATCHL; ++p)
    s += patch[p] * posW[(size_t)p * DMODEL + d];
  h[((size_t)seq * NPAT + t) * DMODEL + d] = s;
}

// ---------------- RMSNorm over D_MODEL ----------------
__global__ void rmsnorm(const float* __restrict__ h, const float* __restrict__ w,
                        float* __restrict__ xn)
{
  const int row = blockIdx.x;
  const int d   = threadIdx.x;
  __shared__ float red[DMODEL];
  float v = h[(size_t)row * DMODEL + d];
  red[d] = v * v;
  __syncthreads();
  for (int s = DMODEL / 2; s > 0; s >>= 1) {
    if (d < s) red[d] += red[d + s];
    __syncthreads();
  }
  float inv = rsqrtf(red[0] / (float)DMODEL + EPSV);
  xn[(size_t)row * DMODEL + d] = v * inv * w[d];
}

// ---------------- fp32 WMMA GEMM: one wave computes a 16 x (16*NB) strip ----------------
// MODE 0: C = A@B    MODE 1: C = softplus(A@B + bias[col])    MODE 2: C += A@B
template<int NB, int MODE>
__global__ void gemm_wmma_f32(const float* __restrict__ A, int lda,
                              const float* __restrict__ B, int ldb,
                              float* __restrict__ C, int ldc,
                              int K, const float* __restrict__ bias)
{
  const int lane = threadIdx.x;        // block = 32 threads = 1 wave32, EXEC all 1s
  const int r    = lane & 15;
  const int hh   = lane >> 4;
  const int m0   = blockIdx.x * 16;
  const int n0   = blockIdx.y * (16 * NB);
  const float* Arow = A + (size_t)(m0 + r) * lda;
  const float* Bcol = B + n0 + r;
  __builtin_prefetch(Arow, 0, 0);      // global_prefetch_b8 (pull A row toward WGP)
  __builtin_prefetch(Bcol, 0, 0);
  v8f zero = {};
  v8f acc[NB];
#pragma unroll
  for (int j = 0; j < NB; ++j) acc[j] = zero;

  for (int k0 = 0; k0 < K; k0 += 4) {
    const int k = k0 + 2 * hh;
    v2f a;
    a[0] = Arow[k];
    a[1] = Arow[k + 1];
    const float* Bk0 = Bcol + (size_t)k * ldb;
    const float* Bk1 = Bcol + (size_t)(k + 1) * ldb;
#pragma unroll
    for (int j = 0; j < NB; ++j) {      // reuse A fragment across NB output tiles
      v2f b;
      b[0] = Bk0[j * 16];
      b[1] = Bk1[j * 16];
      acc[j] = wmma4(a, b, acc[j]);
    }
  }

#pragma unroll
  for (int j = 0; j < NB; ++j) {
    const int nc = n0 + j * 16 + r;
    if (MODE == 0) {
#pragma unroll
      for (int v = 0; v < 8; ++v)
        C[(size_t)(m0 + v + 8 * hh) * ldc + nc] = acc[j][v];
    } else if (MODE == 1) {
      const float bb = bias[nc];
#pragma unroll
      for (int v = 0; v < 8; ++v) {
        float xv = acc[j][v] + bb;
        float sp = (xv > 20.f) ? xv : log1pf(expf(xv));   // softplus
        C[(size_t)(m0 + v + 8 * hh) * ldc + nc] = sp;
      }
    } else {
#pragma unroll
      for (int v = 0; v < 8; ++v) {
        size_t idx = (size_t)(m0 + v + 8 * hh) * ldc + nc;
        C[idx] = acc[j][v] + C[idx];                       // residual add
      }
    }
  }
}

// ---------------- depthwise causal conv (K=4) + bias + SiLU ----------------
__global__ void conv_silu(const float* __restrict__ xz,   // (NROWS, 1024), xi = cols 0..511
                          const float* __restrict__ cW,   // (512, 4)
                          const float* __restrict__ cb,   // (512,)
                          float* __restrict__ xi)
{
  const int row = blockIdx.x;      // seq*64 + t
  const int t   = row & (NPAT - 1);
  const int d   = threadIdx.x;     // 0..511
  float acc = cb[d];
#pragma unroll
  for (int k = 0; k < DCONV; ++k) {
    int tt = t - (DCONV - 1) + k;
    if (tt >= 0)
      acc += cW[d * DCONV + k] * xz[(size_t)(row + tt - t) * 1024 + d];
  }
  xi[(size_t)row * DINNER + d] = silu_f(acc);
}

// ---------------- selective scan ----------------
// 1 block/sequence, 1 lane/channel, 16 states in registers.
// Per-timestep B|C row (32 floats) is the serial-critical load: fetched by the
// Tensor Data Mover into double-buffered LDS one step ahead (tensor_load_to_lds
// + s_wait_tensorcnt), overlapping DMA with the exp/FMA state update.
__global__ void selective_scan(const float* __restrict__ xi,
                               float* __restrict__ dy,          // in: delta; out: gated y
                               const float* __restrict__ dbc,   // (NROWS, 48) : dt|B|C
                               const float* __restrict__ xz,    // z = cols 512..1023
                               const float* __restrict__ A_log, // (512, 16)
                               const float* __restrict__ Dv)    // (512,)
{
  const int seq = blockIdx.x;
  const int d   = threadIdx.x;     // 0..511
  __shared__ float BC[2][2 * DSTATE];   // double-buffered B|C rows

  float A[DSTATE];
#pragma unroll
  for (int s = 0; s < DSTATE; ++s)
    A[s] = -expf(A_log[(size_t)d * DSTATE + s]);
  const float Dd = Dv[d];
  float st[DSTATE];
#pragma unroll
  for (int s = 0; s < DSTATE; ++s) st[s] = 0.f;

  // --- TDM descriptor (D#): 1-row tile of 32 dwords, contiguous copy ---
  const unsigned lds_base = (unsigned)(size_t)(&BC[0][0]);  // low 32b of generic = LDS byte addr
  v8u g1;
  g1[0] = (2u << 16);   // workgroup_mask=0, data_size=2 (4 bytes)
  g1[1] = (32u << 16);  // tensor_dim0[15:0]=32 (bits 63:48 of group1)
  g1[2] = (1u << 16);   // tensor_dim0[31:16]=0, tensor_dim1[15:0]=1
  g1[3] = (32u << 16);  // tensor_dim1[31:16]=0, tile_dim0=32
  g1[4] = 1u;           // tile_dim1=1, tile_dim2=0
  g1[5] = 32u;          // tensor_dim0_stride[31:0]=32
  g1[6] = 0u;           // tensor_dim0_stride[47:32]=0, tensor_dim1_stride lo=0
  g1[7] = 0u;

  auto tdm_issue = [&](int t) {
    size_t gaddr = (size_t)(dbc + ((size_t)seq * NPAT + t) * 48 + DTRANK);
    v4u g0;
    g0[0] = 1u;                                              // count=1, user mode
    g0[1] = lds_base + (unsigned)((t & 1) * (2 * DSTATE * 4));
    g0[2] = (unsigned)(gaddr & 0xFFFFFFFFu);                 // global_addr[31:0]
    g0[3] = (unsigned)((gaddr >> 32) & 0x1FFFFFFu) | (2u << 30); // addr[56:32] | type=2
    asm volatile("tensor_load_to_lds %0, %1" :: "s"(g0), "s"(g1) : "memory");
  };

  if (threadIdx.x < 32) tdm_issue(0);              // prime buffer 0 (wave 0 only)

  for (int t = 0; t < NPAT; ++t) {
    if (threadIdx.x < 32) {                        // wave-uniform branch
      if (t + 1 < NPAT) {
        tdm_issue(t + 1);                          // prefetch next step's B|C
        __builtin_amdgcn_s_wait_tensorcnt((short)1);  // in-order: step t resident
      } else {
        __builtin_amdgcn_s_wait_tensorcnt((short)0);
      }
    }
    __syncthreads();                               // publish buf[t&1] to all waves
    const float* Bc = BC[t & 1];
    const float* Cc = Bc + DSTATE;

    const size_t row = (size_t)seq * NPAT + t;
    const float delta = dy[row * DINNER + d];
    const float u     = xi[row * DINNER + d];
    const float du    = delta * u;
    float y = 0.f;
#pragma unroll
    for (int s = 0; s < DSTATE; ++s) {
      const float dA = expf(delta * A[s]);
      st[s] = dA * st[s] + du * Bc[s];
      y += st[s] * Cc[s];
    }
    y += u * Dd;
    const float z = xz[row * 1024 + DINNER + d];
    dy[row * DINNER + d] = y * silu_f(z);          // in-place over delta (index-matched)
    __syncthreads();                               // all reads done before buf reuse
  }
}

// ---------------- LayerNorm over D_MODEL ----------------
__global__ void layernorm(const float* __restrict__ h, const float* __restrict__ g,
                          const float* __restrict__ b, float* __restrict__ o)
{
  const int row = blockIdx.x;
  const int d   = threadIdx.x;
  __shared__ float red[DMODEL];
  float v = h[(size_t)row * DMODEL + d];
  red[d] = v;
  __syncthreads();
  for (int s = DMODEL / 2; s > 0; s >>= 1) {
    if (d < s) red[d] += red[d + s];
    __syncthreads();
  }
  float mu = red[0] / (float)DMODEL;
  __syncthreads();
  float c = v - mu;
  red[d] = c * c;
  __syncthreads();
  for (int s = DMODEL / 2; s > 0; s >>= 1) {
    if (d < s) red[d] += red[d + s];
    __syncthreads();
  }
  float inv = rsqrtf(red[0] / (float)DMODEL + EPSV);
  o[(size_t)row * DMODEL + d] = c * inv * g[d] + b[d];
}

// ---------------- head GEMM: (112,16384)@(16384,96)+b, 8 waves split K, LDS reduce ----------------
__global__ void head_gemm(const float* __restrict__ A,
                          const float* __restrict__ W,
                          const float* __restrict__ bias,
                          float* __restrict__ out)
{
  __shared__ float tile[8 * 256];
  const int mt = blockIdx.x, nt = blockIdx.y;
  const int wave = threadIdx.x >> 5;
  const int lane = threadIdx.x & 31;
  const int r = lane & 15, hh = lane >> 4;
  const int m0 = mt * 16, n0 = nt * 16;
  const int KTOT = NPAT * DMODEL;   // 16384
  const int KC   = KTOT / 8;        // 2048 per wave
  const float* Arow = A + (size_t)(m0 + r) * KTOT + (size_t)wave * KC;
  const float* Bcol = W + (size_t)wave * KC * PREDL + n0 + r;
  __builtin_prefetch(Arow, 0, 0);
  __builtin_prefetch(Bcol, 0, 0);
  v8f acc = {};
  for (int k0 = 0; k0 < KC; k0 += 4) {
    const int k = k0 + 2 * hh;
    v2f a, b;
    a[0] = Arow[k];
    a[1] = Arow[k + 1];
    b[0] = Bcol[(size_t)k * PREDL];
    b[1] = Bcol[(size_t)(k + 1) * PREDL];
    acc = wmma4(a, b, acc);
  }
#pragma unroll
  for (int v = 0; v < 8; ++v)
    tile[wave * 256 + (v + 8 * hh) * 16 + r] = acc[v];
  __syncthreads();
  const int e = threadIdx.x;        // 0..255 : one output element of the 16x16 tile
  float s = 0.f;
#pragma unroll
  for (int w = 0; w < 8; ++w) s += tile[w * 256 + e];
  const int row = e >> 4, col = e & 15;
  out[(size_t)(m0 + row) * PREDL + n0 + col] = s + bias[n0 + col];
}

extern "C" void kernel_launch(void* const* d_in, const int* in_sizes, int n_in,
                              void* d_out, int out_size, void* d_ws, size_t ws_size,
                              hipStream_t stream)
{
  (void)in_sizes; (void)n_in; (void)out_size; (void)ws_size;
  const float* x          = (const float*)d_in[0];
  const float* pos_W      = (const float*)d_in[1];
  const float* pos_b      = (const float*)d_in[2];
  const float* pos_emb    = (const float*)d_in[3];
  const float* rms_w      = (const float*)d_in[4];
  const float* in_proj_W  = (const float*)d_in[5];
  const float* conv_W     = (const float*)d_in[6];
  const float* conv_b     = (const float*)d_in[7];
  const float* x_proj_W   = (const float*)d_in[8];
  const float* dt_W       = (const float*)d_in[9];
  const float* dt_b       = (const float*)d_in[10];
  const float* A_log      = (const float*)d_in[11];
  const float* D_skip     = (const float*)d_in[12];
  const float* out_proj_W = (const float*)d_in[13];
  const float* ln_g       = (const float*)d_in[14];
  const float* ln_b       = (const float*)d_in[15];
  const float* head_W     = (const float*)d_in[16];
  const float* head_b     = (const float*)d_in[17];
  float* out = (float*)d_out;
  float* ws  = (float*)d_ws;

  // workspace layout (floats)
  float* h_buf   = ws;                                   // 7168*256
  float* xn_buf  = h_buf   + (size_t)NROWS * DMODEL;     // 7168*256
  float* xz_buf  = xn_buf  + (size_t)NROWS * DMODEL;     // 7168*1024
  float* xi_buf  = xz_buf  + (size_t)NROWS * 1024;       // 7168*512
  float* dbc_buf = xi_buf  + (size_t)NROWS * DINNER;     // 7168*48
  float* dl_buf  = dbc_buf + (size_t)NROWS * 48;         // 7168*512 (delta, then y)

  patch_embed<<<dim3(BM_SEQ, NPAT), DMODEL, 0, stream>>>(x, pos_W, pos_b, pos_emb, h_buf);

  for (int l = 0; l < 2; ++l) {
    rmsnorm<<<NROWS, DMODEL, 0, stream>>>(h_buf, rms_w + (size_t)l * DMODEL, xn_buf);

    // xz = xn @ in_proj_W[l]   (7168,256)x(256,1024)
    gemm_wmma_f32<4, 0><<<dim3(NROWS / 16, 1024 / 64), 32, 0, stream>>>(
        xn_buf, DMODEL, in_proj_W + (size_t)l * DMODEL * 1024, 1024,
        xz_buf, 1024, DMODEL, nullptr);

    conv_silu<<<NROWS, DINNER, 0, stream>>>(
        xz_buf, conv_W + (size_t)l * DINNER * DCONV, conv_b + (size_t)l * DINNER, xi_buf);

    // dbc = xi @ x_proj_W[l]   (7168,512)x(512,48)
    gemm_wmma_f32<3, 0><<<dim3(NROWS / 16, 1), 32, 0, stream>>>(
        xi_buf, DINNER, x_proj_W + (size_t)l * DINNER * 48, 48,
        dbc_buf, 48, DINNER, nullptr);

    // delta = softplus(dt @ dt_W[l] + dt_b[l])   (7168,16)x(16,512), A strided lda=48
    gemm_wmma_f32<4, 1><<<dim3(NROWS / 16, DINNER / 64), 32, 0, stream>>>(
        dbc_buf, 48, dt_W + (size_t)l * DTRANK * DINNER, DINNER,
        dl_buf, DINNER, DTRANK, dt_b + (size_t)l * DINNER);

    selective_scan<<<BM_SEQ, DINNER, 0, stream>>>(
        xi_buf, dl_buf, dbc_buf, xz_buf,
        A_log + (size_t)l * DINNER * DSTATE, D_skip + (size_t)l * DINNER);

    // h += y @ out_proj_W[l]   (7168,512)x(512,256)
    gemm_wmma_f32<4, 2><<<dim3(NROWS / 16, DMODEL / 64), 32, 0, stream>>>(
        dl_buf, DINNER, out_proj_W + (size_t)l * DINNER * DMODEL, DMODEL,
        h_buf, DMODEL, DINNER, nullptr);
  }

  layernorm<<<NROWS, DMODEL, 0, stream>>>(h_buf, ln_g, ln_b, xn_buf);

  head_gemm<<<dim3(BM_SEQ / 16, PREDL / 16), 256, 0, stream>>>(xn_buf, head_W, head_b, out);
}